// IDKN_simple_10531259809800
// MI455X (gfx1250) — compile-verified
//
#include <hip/hip_runtime.h>
#include <hip/hip_bf16.h>

typedef __attribute__((ext_vector_type(2))) float v2f;
typedef __attribute__((ext_vector_type(8))) float v8f;

#if defined(__has_builtin)
#if __has_builtin(__builtin_amdgcn_wmma_f32_16x16x4_f32)
#define HAVE_WMMA_F32X4 1
#endif
#endif

#define C_DIM 128

// ---------------------------------------------------------------------------
// CSR construction: degree count -> exclusive scan -> bin edges by dst
// ---------------------------------------------------------------------------
__global__ __launch_bounds__(256) void zero_int_kernel(int* p, int n) {
    int i = blockIdx.x * blockDim.x + threadIdx.x;
    if (i < n) p[i] = 0;
}

__global__ __launch_bounds__(256) void deg_count_kernel(const long long* __restrict__ dst,
                                                        int* degcnt, int n_edges) {
    int e = blockIdx.x * blockDim.x + threadIdx.x;
    if (e < n_edges) atomicAdd(&degcnt[(int)dst[e]], 1);
}

// dinv[i] = rsqrt(deg_in[i] + 1)   (+1 = self-loop; always > 0)
__global__ __launch_bounds__(256) void dinv_kernel(const int* __restrict__ degcnt,
                                                   float* dinv, int n) {
    int i = blockIdx.x * blockDim.x + threadIdx.x;
    if (i < n) dinv[i] = rsqrtf((float)degcnt[i] + 1.0f);
}

// Single-workgroup exclusive scan (Hillis-Steele per 256-chunk with carry).
__global__ __launch_bounds__(256) void scan_kernel(const int* __restrict__ degcnt,
                                                   int* __restrict__ rowstart, int n) {
    __shared__ int smem[256];
    __shared__ int carry_s;
    if (threadIdx.x == 0) carry_s = 0;
    __syncthreads();
    for (int base = 0; base < n; base += 256) {
        int i = base + (int)threadIdx.x;
        int v = (i < n) ? degcnt[i] : 0;
        smem[threadIdx.x] = v;
        __syncthreads();
        for (int off = 1; off < 256; off <<= 1) {
            int t = (threadIdx.x >= (unsigned)off) ? smem[threadIdx.x - off] : 0;
            __syncthreads();
            smem[threadIdx.x] += t;
            __syncthreads();
        }
        int incl = smem[threadIdx.x];
        int c = carry_s;
        if (i < n) rowstart[i] = c + incl - v;
        __syncthreads();
        if (threadIdx.x == 0) carry_s = c + smem[255];
        __syncthreads();
    }
    if (threadIdx.x == 0) rowstart[n] = carry_s;
}

__global__ __launch_bounds__(256) void copy_int_kernel(const int* __restrict__ a,
                                                       int* b, int n) {
    int i = blockIdx.x * blockDim.x + threadIdx.x;
    if (i < n) b[i] = a[i];
}

__global__ __launch_bounds__(256) void bin_kernel(const long long* __restrict__ src,
                                                  const long long* __restrict__ dst,
                                                  int* cursor, int* __restrict__ csr_src,
                                                  int n_edges) {
    int e = blockIdx.x * blockDim.x + threadIdx.x;
    if (e < n_edges) {
        int d = (int)dst[e];
        int pos = atomicAdd(&cursor[d], 1);
        csr_src[pos] = (int)src[e];
    }
}

// ---------------------------------------------------------------------------
// GEMM: out[N,128] = (RELU ? relu(in) : in) @ W[128,128]
// One wave computes one 16x16 tile via V_WMMA_F32_16X16X4_F32 (K-step 4).
// Block = 256 threads = 8 waves -> 8 column tiles (covers all 128 cols).
// ---------------------------------------------------------------------------
template <bool RELU>
__global__ __launch_bounds__(256) void gemm128_kernel(const float* __restrict__ in,
                                                      const float* __restrict__ W,
                                                      float* __restrict__ out,
                                                      int n_rows) {
    const int lane = threadIdx.x & 31;
    const int wave = threadIdx.x >> 5;      // 0..7 -> column tile
    const int rowBase = blockIdx.x * 16;
    const int colBase = wave * 16;
    const int m  = lane & 15;
    const int hi = lane >> 4;               // 0: K pair {0,1}, 1: K pair {2,3}

    int arow = rowBase + m;
    if (arow >= n_rows) arow = n_rows - 1;  // clamp loads; EXEC stays all-ones
    const int bcol = colBase + m;

#ifdef HAVE_WMMA_F32X4
    v8f acc = {};
    for (int k = 0; k < C_DIM; k += 4) {
        // A 16x4: lanes 0-15 hold K=k,k+1; lanes 16-31 hold K=k+2,k+3
        const float* ap = in + (size_t)arow * C_DIM + k + hi * 2;
        v2f a;
        a.x = ap[0];
        a.y = ap[1];
        if (RELU) { a.x = fmaxf(a.x, 0.0f); a.y = fmaxf(a.y, 0.0f); }
        // B 4x16: rows striped across lanes within a VGPR; K halves by lane half
        const float* bp = W + (size_t)(k + hi * 2) * C_DIM + bcol;
        v2f b;
        b.x = bp[0];
        b.y = bp[C_DIM];
        acc = __builtin_amdgcn_wmma_f32_16x16x4_f32(
            /*neg_a=*/false, a, /*neg_b=*/false, b,
            /*c_mod=*/(short)0, acc, /*reuse_a=*/false, /*reuse_b=*/false);
    }
#else
    float accv[8] = {};
    for (int r = 0; r < 8; ++r) {
        int rr = rowBase + hi * 8 + r;
        if (rr >= n_rows) rr = n_rows - 1;
        const float* ap = in + (size_t)rr * C_DIM;
        float s = 0.0f;
        for (int k = 0; k < C_DIM; ++k) {
            float av = ap[k];
            if (RELU) av = fmaxf(av, 0.0f);
            s += av * W[(size_t)k * C_DIM + bcol];
        }
        accv[r] = s;
    }
    v8f acc;
    for (int r = 0; r < 8; ++r) acc[r] = accv[r];
#endif

    // D layout: VGPR r -> row (rowBase + hi*8 + r), col (colBase + lane&15)
    const int srow = rowBase + hi * 8;
    float* op = out + (size_t)srow * C_DIM + colBase + m;
#pragma unroll
    for (int r = 0; r < 8; ++r) {
        if (srow + r < n_rows) op[(size_t)r * C_DIM] = acc[r];
    }
}

// ---------------------------------------------------------------------------
// Per-node aggregation (no atomics): one wave per node, lanes hold float4 of
// the 128 channels. acc = bias + dinv[n]^2*h[n] + sum_j dinv[n]*dinv[s_j]*h[s_j]
// Each edge iteration is one coalesced 512B gather + register FMA.
// ---------------------------------------------------------------------------
__global__ __launch_bounds__(256) void node_agg_kernel(const float* __restrict__ h,
                                                       const float* __restrict__ dinv,
                                                       const float* __restrict__ bias,
                                                       const int* __restrict__ rowstart,
                                                       const int* __restrict__ csr_src,
                                                       float* __restrict__ agg,
                                                       int n_nodes) {
    int idx  = blockIdx.x * blockDim.x + threadIdx.x;
    int node = idx >> 5;
    int lane = idx & 31;
    if (node >= n_nodes) return;

    float dn = dinv[node];
    float4 hv = ((const float4*)(h + (size_t)node * C_DIM))[lane];
    float4 bv = ((const float4*)bias)[lane];
    float w0 = dn * dn;
    float4 acc;
    acc.x = bv.x + w0 * hv.x;
    acc.y = bv.y + w0 * hv.y;
    acc.z = bv.z + w0 * hv.z;
    acc.w = bv.w + w0 * hv.w;

    const int s0 = rowstart[node];
    const int s1 = rowstart[node + 1];
    for (int j = s0; j < s1; ++j) {
        int s = csr_src[j];
        float w = dn * dinv[s];
        float4 mv = ((const float4*)(h + (size_t)s * C_DIM))[lane];
        acc.x += w * mv.x;
        acc.y += w * mv.y;
        acc.z += w * mv.z;
        acc.w += w * mv.w;
    }
    ((float4*)(agg + (size_t)node * C_DIM))[lane] = acc;
}

// ---------------------------------------------------------------------------
// out[i] = relu(agg[i,:]) . Wlin + blin   (wave-per-node dot + shuffle reduce)
// ---------------------------------------------------------------------------
__global__ __launch_bounds__(256) void final_kernel(const float* __restrict__ agg,
                                                    const float* __restrict__ Wlin,
                                                    const float* __restrict__ blin,
                                                    float* __restrict__ out,
                                                    int n_nodes) {
    int idx  = blockIdx.x * blockDim.x + threadIdx.x;
    int node = idx >> 5;
    int lane = idx & 31;
    if (node >= n_nodes) return;
    float4 hv = ((const float4*)(agg + (size_t)node * C_DIM))[lane];
    float4 wv = ((const float4*)Wlin)[lane];
    float p = fmaxf(hv.x, 0.0f) * wv.x + fmaxf(hv.y, 0.0f) * wv.y +
              fmaxf(hv.z, 0.0f) * wv.z + fmaxf(hv.w, 0.0f) * wv.w;
#pragma unroll
    for (int off = 16; off > 0; off >>= 1) p += __shfl_down(p, off, 32);
    if (lane == 0) out[node] = p + blin[0];
}

// ---------------------------------------------------------------------------
extern "C" void kernel_launch(void* const* d_in, const int* in_sizes, int n_in,
                              void* d_out, int out_size, void* d_ws, size_t ws_size,
                              hipStream_t stream) {
    const float*     x    = (const float*)d_in[0];
    // d_in[1] = x2 (unused by reference)
    const long long* edge = (const long long*)d_in[2];
    // d_in[3] = num_nodes scalar (derive instead)
    const float* W1   = (const float*)d_in[4];
    const float* b1   = (const float*)d_in[5];
    const float* W2   = (const float*)d_in[6];
    const float* b2   = (const float*)d_in[7];
    const float* Wlin = (const float*)d_in[8];
    const float* blin = (const float*)d_in[9];

    const int n_nodes = in_sizes[0] / C_DIM;   // 50000
    const int n_edges = in_sizes[2] / 2;       // 800000
    const long long* src = edge;
    const long long* dst = edge + n_edges;

    // Workspace carve-out:
    //  dinv[npad] | rowstart[npad] | cursor[npad] | csr_src[epad] | bufA | bufB
    size_t npad = (((size_t)n_nodes + 1 + 255) / 256) * 256;
    size_t epad = (((size_t)n_edges + 255) / 256) * 256;
    float* dinv     = (float*)d_ws;
    int*   rowstart = (int*)(dinv + npad);
    int*   cursor   = rowstart + npad;
    int*   csr_src  = cursor + npad;
    float* bufA     = (float*)(csr_src + epad);
    float* bufB     = bufA + npad * C_DIM;

    const int TPB = 256;
    const int nodeBlocks     = (n_nodes + TPB - 1) / TPB;
    const int edgeBlocks     = (n_edges + TPB - 1) / TPB;
    const int nodeLaneBlocks = (int)(((size_t)n_nodes * 32 + TPB - 1) / TPB);
    const int gemmBlocks     = (n_nodes + 15) / 16;

    // ---- CSR build + normalization (no float atomics anywhere) ----
    zero_int_kernel<<<nodeBlocks, TPB, 0, stream>>>(cursor, n_nodes);          // cursor = degcnt
    deg_count_kernel<<<edgeBlocks, TPB, 0, stream>>>(dst, cursor, n_edges);
    dinv_kernel<<<nodeBlocks, TPB, 0, stream>>>(cursor, dinv, n_nodes);
    scan_kernel<<<1, TPB, 0, stream>>>(cursor, rowstart, n_nodes);
    copy_int_kernel<<<nodeBlocks, TPB, 0, stream>>>(rowstart, cursor, n_nodes);
    bin_kernel<<<edgeBlocks, TPB, 0, stream>>>(src, dst, cursor, csr_src, n_edges);

    // ---- Layer 1: h1 = x @ W1 ; agg1 = b1 + D^-1/2 A D^-1/2 h1 ----
    gemm128_kernel<false><<<gemmBlocks, 256, 0, stream>>>(x, W1, bufA, n_nodes);
    node_agg_kernel<<<nodeLaneBlocks, TPB, 0, stream>>>(bufA, dinv, b1, rowstart,
                                                        csr_src, bufB, n_nodes);

    // ---- Layer 2: h2 = relu(agg1) @ W2 ; agg2 = b2 + aggregation ----
    gemm128_kernel<true><<<gemmBlocks, 256, 0, stream>>>(bufB, W2, bufA, n_nodes);
    node_agg_kernel<<<nodeLaneBlocks, TPB, 0, stream>>>(bufA, dinv, b2, rowstart,
                                                        csr_src, bufB, n_nodes);

    // ---- Output head: out = relu(agg2) @ Wlin + blin ----
    final_kernel<<<nodeLaneBlocks, TPB, 0, stream>>>(bufB, Wlin, blin,
                                                     (float*)d_out, n_nodes);
}